// GATNet_146028888429
// MI455X (gfx1250) — compile-verified
//
#include <hip/hip_runtime.h>
#include <hip/hip_bf16.h>
#include <math.h>

#define N_NODES 30000
#define N_EDGES 480000
#define NGRAPH  32
#define CDIV(a,b) (((a)+(b)-1)/(b))

typedef float v2f __attribute__((ext_vector_type(2)));
typedef float v8f __attribute__((ext_vector_type(8)));

// ---------- order-preserving float<->uint for atomicMax on floats ----------
__device__ __forceinline__ unsigned f32_order(float f) {
    unsigned u = __float_as_uint(f);
    return (u & 0x80000000u) ? ~u : (u | 0x80000000u);
}
__device__ __forceinline__ float f32_unorder(unsigned u) {
    u = (u & 0x80000000u) ? (u & 0x7fffffffu) : ~u;
    return __uint_as_float(u);
}

// ---------- generic fill ----------
__global__ void fill_f32_k(float* __restrict__ p, float v, size_t n) {
    size_t i = (size_t)blockIdx.x * blockDim.x + threadIdx.x;
    if (i < n) p[i] = v;
}

// ---------- degree + edge-attr sum (for fill_value='mean' self loops) ----------
__global__ void deg_easum_k(const int* __restrict__ dst, const float* __restrict__ ea,
                            float* __restrict__ deg, float* __restrict__ easum, int E) {
    int e = blockIdx.x * blockDim.x + threadIdx.x;
    if (e >= E) return;
    int t = dst[e];
    atomicAdd(&deg[t], 1.0f);
    atomicAdd(&easum[t], ea[e]);
}
__global__ void eamean_k(float* __restrict__ eamean, const float* __restrict__ deg, int n) {
    int i = blockIdx.x * blockDim.x + threadIdx.x;
    if (i < n) eamean[i] = eamean[i] / fmaxf(deg[i], 1.0f);  // in-place: eamean held the sum
}

// ---------- fp32 WMMA GEMM: C[N,M] = A[N,K] @ B[K,M], one 16x16 tile per wave ----------
// A frag (16x4 f32): lane<16 -> row=lane, K={k0,k0+1}; lane>=16 -> row=lane-16, K={k0+2,k0+3}
// B frag (4x16 f32): mirrored over columns. C/D: VGPR v -> row v + 8*(lane/16), col lane%16.
// Branch-free inner loop: main K4 loop unguarded; M-edge handled with clamped
// addresses + per-lane selects (v_cndmask), never EXEC manipulation; K tail = 1 extra WMMA.
__global__ void gemm16_wmma_k(const float* __restrict__ A, const float* __restrict__ B,
                              float* __restrict__ C, int K, int M, int mtiles, int totalTiles) {
    int wave = blockIdx.x * (blockDim.x >> 5) + (threadIdx.x >> 5);
    if (wave >= totalTiles) return;                 // wave-uniform: EXEC stays all-ones
    int tr = wave / mtiles, tc = wave % mtiles;
    int lane = threadIdx.x & 31;
    int half = lane >> 4;
    int l16  = lane & 15;
    int row = tr * 16 + l16;
    int col = tc * 16 + l16;
    bool cin = (col < M);
    int colc = cin ? col : (M - 1);                 // always a valid column to load
    const float* Ap = A + (size_t)row * K + half * 2;
    const float* Bp = B + (size_t)(half * 2) * M + colc;
    v8f acc = {0.f, 0.f, 0.f, 0.f, 0.f, 0.f, 0.f, 0.f};
    const int K4 = K & ~3;
    for (int k0 = 0; k0 < K4; k0 += 4) {
        v2f a, b;
        a.x = Ap[0];
        a.y = Ap[1];
        float b0 = Bp[0];
        float b1 = Bp[M];
        b.x = cin ? b0 : 0.0f;
        b.y = cin ? b1 : 0.0f;
        acc = __builtin_amdgcn_wmma_f32_16x16x4_f32(false, a, false, b, (short)0, acc,
                                                    false, false);
        Ap += 4;
        Bp += (size_t)4 * M;
    }
    if (K4 < K) {                                   // K tail (K%4 != 0): selects, no branches
        const float* Arow = A + (size_t)row * K;
        int ka  = K4 + half * 2;
        int k0c = (ka     < K) ? ka     : (K - 1);
        int k1c = (ka + 1 < K) ? ka + 1 : (K - 1);
        v2f a, b;
        float a0 = Arow[k0c];
        float a1 = Arow[k1c];
        a.x = (ka     < K) ? a0 : 0.0f;
        a.y = (ka + 1 < K) ? a1 : 0.0f;
        float b0 = B[(size_t)k0c * M + colc];
        float b1 = B[(size_t)k1c * M + colc];
        b.x = (cin && ka     < K) ? b0 : 0.0f;
        b.y = (cin && ka + 1 < K) ? b1 : 0.0f;
        acc = __builtin_amdgcn_wmma_f32_16x16x4_f32(false, a, false, b, (short)0, acc,
                                                    false, false);
    }
    if (cin) {
#pragma unroll
        for (int v = 0; v < 8; ++v) {
            int r = tr * 16 + v + half * 8;
            C[(size_t)r * M + col] = acc[v];
        }
    }
}

// ---------- per-(n,h) attention dots: a_s = <xw[n,h,:], a_src[h,:]>, a_d likewise ----------
__global__ void attn_coef_k(const float* __restrict__ xw, const float* __restrict__ a_src,
                            const float* __restrict__ a_dst, float* __restrict__ as_,
                            float* __restrict__ ad_, int total, int H, int C) {
    int idx = blockIdx.x * blockDim.x + threadIdx.x;
    if (idx >= total) return;                       // total = N*H, idx = n*H+h
    int h = idx % H;
    const float* xp = xw + (size_t)idx * C;
    const float* ps = a_src + h * C;
    const float* pd = a_dst + h * C;
    float ss = 0.f, sd = 0.f;
    for (int c = 0; c < C; ++c) { float v = xp[c]; ss += v * ps[c]; sd += v * pd[c]; }
    as_[idx] = ss; ad_[idx] = sd;
}

// ---------- edge-attr attention coefficient per head (edge_dim==1) ----------
__global__ void ecoef_k(const float* __restrict__ We, const float* __restrict__ a_e,
                        float* __restrict__ ecoef, int H, int C) {
    int h = threadIdx.x;
    if (h >= H) return;
    float s = 0.f;
    for (int c = 0; c < C; ++c) s += We[h * C + c] * a_e[h * C + c];
    ecoef[h] = s;
}

// ---------- self-loop logits; seed segment max ----------
__global__ void self_alpha_k(const float* __restrict__ as_, const float* __restrict__ ad_,
                             const float* __restrict__ ecoef, const float* __restrict__ eamean,
                             float* __restrict__ aself, unsigned* __restrict__ amax,
                             int total, int H) {
    int idx = blockIdx.x * blockDim.x + threadIdx.x;
    if (idx >= total) return;                       // idx = n*H+h
    int n = idx / H, h = idx % H;
    float ap = as_[idx] + ad_[idx] + ecoef[h] * eamean[n];
    ap = (ap > 0.f) ? ap : 0.2f * ap;               // leaky_relu(0.2)
    aself[idx] = ap;
    amax[idx]  = f32_order(ap);
}

// ---------- real-edge logits + atomic segment max ----------
__global__ void edge_alpha_k(const int* __restrict__ src, const int* __restrict__ dst,
                             const float* __restrict__ ea, const float* __restrict__ as_,
                             const float* __restrict__ ad_, const float* __restrict__ ecoef,
                             float* __restrict__ epre, unsigned* __restrict__ amax,
                             int total, int H) {
    int idx = blockIdx.x * blockDim.x + threadIdx.x;
    if (idx >= total) return;                       // total = E*H
    int e = idx / H, h = idx % H;
    int s = src[e], t = dst[e];
    float ap = as_[s * H + h] + ad_[t * H + h] + ecoef[h] * ea[e];
    ap = (ap > 0.f) ? ap : 0.2f * ap;
    epre[idx] = ap;
    atomicMax(&amax[t * H + h], f32_order(ap));
}

// ---------- denom seeded with self-loop exp ----------
__global__ void denom_init_k(const float* __restrict__ aself, const unsigned* __restrict__ amax,
                             float* __restrict__ denom, int total) {
    int idx = blockIdx.x * blockDim.x + threadIdx.x;
    if (idx >= total) return;
    denom[idx] = expf(aself[idx] - f32_unorder(amax[idx]));
}

// ---------- accumulator seeded with self-loop contribution ----------
__global__ void acc_init_k(const float* __restrict__ xw, const float* __restrict__ denom,
                           float* __restrict__ acc, size_t total, int H, int C) {
    size_t idx = (size_t)blockIdx.x * blockDim.x + threadIdx.x;
    if (idx >= total) return;                       // total = N*H*C
    int M = H * C;
    int h = (int)((idx % M) / C);
    size_t n = idx / M;
    acc[idx] = xw[idx] * denom[n * H + h];          // denom == exp(self - max) at this point
}

// ---------- per-edge exp weights + denom atomics (epre -> weights in place) ----------
__global__ void ew_k(const int* __restrict__ dst, float* __restrict__ epre,
                     const unsigned* __restrict__ amax, float* __restrict__ denom,
                     int total, int H) {
    int idx = blockIdx.x * blockDim.x + threadIdx.x;
    if (idx >= total) return;                       // total = E*H
    int e = idx / H, h = idx % H;
    int t = dst[e];
    float w = expf(epre[idx] - f32_unorder(amax[t * H + h]));
    epre[idx] = w;
    atomicAdd(&denom[t * H + h], w);
}

// ---------- edge scatter: acc[t,h,c] += xw[s,h,c] * w[e,h] ----------
__global__ void scatter_k(const float* __restrict__ xw, const float* __restrict__ ew,
                          const int* __restrict__ src, const int* __restrict__ dst,
                          float* __restrict__ acc, size_t total, int H, int C) {
    size_t idx = (size_t)blockIdx.x * blockDim.x + threadIdx.x;
    if (idx >= total) return;                       // total = E*H*C
    int M = H * C;
    size_t e = idx / M;
    int m = (int)(idx % M);
    int h = m / C;
    int s = src[e], t = dst[e];
    float w = ew[e * H + h];
    atomicAdd(&acc[(size_t)t * M + m], xw[(size_t)s * M + m] * w);
}

// ---------- finalize: concat heads + bias + ELU ----------
__global__ void finalize_concat_k(const float* __restrict__ acc, const float* __restrict__ denom,
                                  const float* __restrict__ bias, float* __restrict__ out,
                                  size_t total, int H, int C) {
    size_t idx = (size_t)blockIdx.x * blockDim.x + threadIdx.x;
    if (idx >= total) return;                       // total = N*H*C
    int M = H * C;
    int m = (int)(idx % M);
    int h = m / C;
    size_t n = idx / M;
    float y = acc[idx] / denom[n * H + h] + bias[m];
    out[idx] = (y > 0.f) ? y : expm1f(y);           // ELU(alpha=1)
}

// ---------- finalize: mean over heads + bias (no activation) ----------
__global__ void finalize_mean_k(const float* __restrict__ acc, const float* __restrict__ denom,
                                const float* __restrict__ bias, float* __restrict__ out,
                                int total, int H, int C) {
    int idx = blockIdx.x * blockDim.x + threadIdx.x;
    if (idx >= total) return;                       // total = N*C
    int n = idx / C, c = idx % C;
    float s = 0.f;
    for (int h = 0; h < H; ++h) s += acc[((size_t)n * H + h) * C + c] / denom[n * H + h];
    out[idx] = s / (float)H + bias[c];
}

// ---------- final FC over last node of each (equal-sized, sorted) graph ----------
__global__ void final_fc_k(const float* __restrict__ h2, const float* __restrict__ p2,
                           const float* __restrict__ Wfc, const float* __restrict__ bfc,
                           float* __restrict__ out) {
    int g = blockIdx.x;
    int j = threadIdx.x;
    if (j >= 129) return;
    int last = ((g + 1) * N_NODES - 1) / NGRAPH;    // last node index of graph g
    float s = bfc[j];
    for (int k = 0; k < 129; ++k) s += h2[(size_t)last * 129 + k] * Wfc[k * 129 + j];
    for (int k = 0; k < 129; ++k) s += p2[(size_t)last * 129 + k] * Wfc[(129 + k) * 129 + j];
    out[g * 129 + j] = s;
}

// ===================== host-side orchestration =====================
static void run_gat_layer(hipStream_t stream,
                          const float* x, int K,
                          const int* src, const int* dst, const float* ea,
                          const float* W, const float* a_src, const float* a_dst,
                          const float* We, const float* a_e, const float* bias,
                          int H, int C, bool concat_elu, float* out,
                          float* xw, float* acc, float* as_, float* ad_, float* aself,
                          unsigned* amax, float* denom, float* epre,
                          const float* eamean, float* ecoef) {
    const int N = N_NODES, E = N_EDGES;
    const int M = H * C;
    const int ntiles = N / 16, mtiles = CDIV(M, 16);
    const int totalTiles = ntiles * mtiles;
    const int WPB = 4;                               // waves per block
    gemm16_wmma_k<<<CDIV(totalTiles, WPB), 32 * WPB, 0, stream>>>(x, W, xw, K, M, mtiles,
                                                                  totalTiles);
    const int nh = N * H;
    attn_coef_k<<<CDIV(nh, 256), 256, 0, stream>>>(xw, a_src, a_dst, as_, ad_, nh, H, C);
    ecoef_k<<<1, 32, 0, stream>>>(We, a_e, ecoef, H, C);
    self_alpha_k<<<CDIV(nh, 256), 256, 0, stream>>>(as_, ad_, ecoef, eamean, aself, amax, nh, H);
    const int eh = E * H;
    edge_alpha_k<<<CDIV(eh, 256), 256, 0, stream>>>(src, dst, ea, as_, ad_, ecoef, epre, amax,
                                                    eh, H);
    denom_init_k<<<CDIV(nh, 256), 256, 0, stream>>>(aself, amax, denom, nh);
    const size_t nhc = (size_t)N * M;
    acc_init_k<<<(unsigned)CDIV(nhc, (size_t)256), 256, 0, stream>>>(xw, denom, acc, nhc, H, C);
    ew_k<<<CDIV(eh, 256), 256, 0, stream>>>(dst, epre, amax, denom, eh, H);
    const size_t ehc = (size_t)E * M;
    scatter_k<<<(unsigned)CDIV(ehc, (size_t)256), 256, 0, stream>>>(xw, epre, src, dst, acc,
                                                                    ehc, H, C);
    if (concat_elu) {
        finalize_concat_k<<<(unsigned)CDIV(nhc, (size_t)256), 256, 0, stream>>>(acc, denom, bias,
                                                                                out, nhc, H, C);
    } else {
        finalize_mean_k<<<CDIV(N * C, 256), 256, 0, stream>>>(acc, denom, bias, out, N * C, H, C);
    }
}

extern "C" void kernel_launch(void* const* d_in, const int* in_sizes, int n_in,
                              void* d_out, int out_size, void* d_ws, size_t ws_size,
                              hipStream_t stream) {
    const int N = N_NODES, E = N_EDGES;
    const float* x_hrtf = (const float*)d_in[0];
    const float* x_pos  = (const float*)d_in[1];
    const float* ea_h   = (const float*)d_in[2];
    const float* ea_p   = (const float*)d_in[3];
    const float* W1h = (const float*)d_in[4],  *as1h = (const float*)d_in[5];
    const float* ad1h = (const float*)d_in[6], *We1h = (const float*)d_in[7];
    const float* ae1h = (const float*)d_in[8], *b1h  = (const float*)d_in[9];
    const float* W2h = (const float*)d_in[10], *as2h = (const float*)d_in[11];
    const float* ad2h = (const float*)d_in[12], *We2h = (const float*)d_in[13];
    const float* ae2h = (const float*)d_in[14], *b2h  = (const float*)d_in[15];
    const float* W1p = (const float*)d_in[16], *as1p = (const float*)d_in[17];
    const float* ad1p = (const float*)d_in[18], *We1p = (const float*)d_in[19];
    const float* ae1p = (const float*)d_in[20], *b1p  = (const float*)d_in[21];
    const float* W2p = (const float*)d_in[22], *as2p = (const float*)d_in[23];
    const float* ad2p = (const float*)d_in[24], *We2p = (const float*)d_in[25];
    const float* ae2p = (const float*)d_in[26], *b2p  = (const float*)d_in[27];
    const float* Wfc = (const float*)d_in[28], *bfc = (const float*)d_in[29];
    const int* ei_h = (const int*)d_in[30];
    const int* ei_p = (const int*)d_in[31];
    const int *src_h = ei_h, *dst_h = ei_h + E;
    const int *src_p = ei_p, *dst_p = ei_p + E;

    // ---- workspace carve (peak ~236 MB) ----
    char* w = (char*)d_ws;
    auto take = [&](size_t bytes) -> char* {
        char* p = w;
        w += (bytes + 255) & ~(size_t)255;
        return p;
    };
    float*    xw     = (float*)take((size_t)N * 512 * 4);
    float*    acc    = (float*)take((size_t)N * 512 * 4);
    float*    h1     = (float*)take((size_t)N * 512 * 4);
    float*    outh   = (float*)take((size_t)N * 129 * 4);
    float*    outp   = (float*)take((size_t)N * 129 * 4);
    float*    as_    = (float*)take((size_t)N * 8 * 4);
    float*    ad_    = (float*)take((size_t)N * 8 * 4);
    float*    aself  = (float*)take((size_t)N * 8 * 4);
    unsigned* amax   = (unsigned*)take((size_t)N * 8 * 4);
    float*    denom  = (float*)take((size_t)N * 8 * 4);
    float*    epre   = (float*)take((size_t)E * 8 * 4);
    float*    eamean = (float*)take((size_t)N * 4);
    float*    deg    = (float*)take((size_t)N * 4);
    float*    ecoef  = (float*)take(256);

    // ---------------- hrtf branch ----------------
    fill_f32_k<<<CDIV(N, 256), 256, 0, stream>>>(deg, 0.f, (size_t)N);
    fill_f32_k<<<CDIV(N, 256), 256, 0, stream>>>(eamean, 0.f, (size_t)N);
    deg_easum_k<<<CDIV(E, 256), 256, 0, stream>>>(dst_h, ea_h, deg, eamean, E);
    eamean_k<<<CDIV(N, 256), 256, 0, stream>>>(eamean, deg, N);
    run_gat_layer(stream, x_hrtf, 129, src_h, dst_h, ea_h, W1h, as1h, ad1h, We1h, ae1h, b1h,
                  8, 64, true, h1, xw, acc, as_, ad_, aself, amax, denom, epre, eamean, ecoef);
    run_gat_layer(stream, h1, 512, src_h, dst_h, ea_h, W2h, as2h, ad2h, We2h, ae2h, b2h,
                  1, 129, false, outh, xw, acc, as_, ad_, aself, amax, denom, epre, eamean,
                  ecoef);

    // ---------------- pos branch ----------------
    fill_f32_k<<<CDIV(N, 256), 256, 0, stream>>>(deg, 0.f, (size_t)N);
    fill_f32_k<<<CDIV(N, 256), 256, 0, stream>>>(eamean, 0.f, (size_t)N);
    deg_easum_k<<<CDIV(E, 256), 256, 0, stream>>>(dst_p, ea_p, deg, eamean, E);
    eamean_k<<<CDIV(N, 256), 256, 0, stream>>>(eamean, deg, N);
    run_gat_layer(stream, x_pos, 15, src_p, dst_p, ea_p, W1p, as1p, ad1p, We1p, ae1p, b1p,
                  8, 64, true, h1, xw, acc, as_, ad_, aself, amax, denom, epre, eamean, ecoef);
    run_gat_layer(stream, h1, 512, src_p, dst_p, ea_p, W2p, as2p, ad2p, We2p, ae2p, b2p,
                  1, 129, false, outp, xw, acc, as_, ad_, aself, amax, denom, epre, eamean,
                  ecoef);

    // ---------------- readout + FC ----------------
    final_fc_k<<<NGRAPH, 160, 0, stream>>>(outh, outp, Wfc, bfc, (float*)d_out);
}